// JastrowNet_39771397160975
// MI455X (gfx1250) — compile-verified
//
#include <hip/hip_runtime.h>
#include <hip/hip_bf16.h>

typedef __attribute__((ext_vector_type(2))) float v2f;
typedef __attribute__((ext_vector_type(8))) float v8f;

#define LN2F  0.69314718055994531f
#define L2EF  1.44269504088896341f

// ---------------- packed weight image layout (float offsets) ----------------
#define L_STRIDE   2048
#define OFF_BW1E   0     // 4x16  [k][n]  n<8: wW1[l,0], n>=8: wW1[l,1]
#define OFF_B1E    64    // 16    wb1[l,0]|wb1[l,1]
#define OFF_BW2E   80    // 16x16 blockdiag(wW2[l,0], wW2[l,1])
#define OFF_B2E    336   // 16    bias, pre-shifted by -ln2*colsum(BW2E)
#define OFF_BW1N   352   // 4x16  wW1[l,2] | 0
#define OFF_B1N    416   // 16
#define OFF_BW2N   432   // 16x16 wW2[l,2] top-left 8x8, else 0
#define OFF_B2N    688   // 16    bias, pre-shifted
#define OFF_BHW    704   // 16x16 hW[l] duplicated cols (n&7)
#define OFF_BHB    960   // 16    hb[l] duplicated
#define OFF_BG1    976   // 16x16 blockdiag(gW1[l,0], gW1[l,1])
#define OFF_BG1B   1232  // 16
#define OFF_BG2    1248  // 16x16 vstack(gW2[l,0], gW2[l,1])
#define OFF_BG2B   1504  // 16    sum gb2 - ln2*(colsum BG2 + colsum BG2N)
#define OFF_BG1N   1520  // 16x16 gW1[l,2] top-left 8x8, else 0
#define OFF_BG1NB  1776  // 16
#define OFF_BG2N   1792  // 16x16 rows0-7 = gW2[l,2], rows8-15 = 0
#define OFF_Y      4096  // 4x8
#define OFF_XINIT  4128  // 16x16
#define OFF_OW     4384  // 16
#define OFF_OB     4400  // 1
#define PACK_TOTAL 4401
#define PACK_PAD   4416

#define SCRW 18          // LDS stage-row stride (floats): 8B aligned, conflict-free
#define SCR_REGION (16*SCRW)

// ---------------- device helpers ----------------
__device__ __forceinline__ v8f wmma4(v2f a, v2f b, v8f c) {
  // D = A(16x4,f32) * B(4x16,f32) + C(16x16,f32) : V_WMMA_F32_16X16X4_F32
  return __builtin_amdgcn_wmma_f32_16x16x4_f32(false, a, false, b, (short)0, c,
                                               false, false);
}
__device__ __forceinline__ v2f load_a4(const float* S, int kofs, int half, int m) {
  return *reinterpret_cast<const v2f*>(S + m * SCRW + kofs + 2 * half);
}
__device__ __forceinline__ v2f load_b4(const float* Bm, int kofs, int half, int n) {
  const float* p = Bm + (kofs + 2 * half) * 16 + n;
  v2f b; b.x = p[0]; b.y = p[16]; return b;
}
__device__ __forceinline__ void store_d(float* S, v8f d, int half, int n) {
#pragma unroll
  for (int v = 0; v < 8; ++v) S[(v + 8 * half) * SCRW + n] = d[v];
}
__device__ __forceinline__ v8f bias8(float bv) {
  v8f c;
#pragma unroll
  for (int v = 0; v < 8; ++v) c[v] = bv;
  return c;
}

// softplus(x) = max(x,0) + ln2*log2(1 + exp2(-|x|*log2e))
// (the reference's -ln2 shift is folded into the downstream packed biases)
// raw v_exp_f32/v_log_f32: log input is in [1,2] -> no denorm guards needed.
// trailing v_nop satisfies the TRANS->consumer 1-instruction gap rule.
__device__ __forceinline__ v8f sp8(v8f x) {
  float m0 = -fabsf(x[0]) * L2EF, m1 = -fabsf(x[1]) * L2EF;
  float m2 = -fabsf(x[2]) * L2EF, m3 = -fabsf(x[3]) * L2EF;
  float m4 = -fabsf(x[4]) * L2EF, m5 = -fabsf(x[5]) * L2EF;
  float m6 = -fabsf(x[6]) * L2EF, m7 = -fabsf(x[7]) * L2EF;
  float t0, t1, t2, t3, t4, t5, t6, t7;
  asm("v_exp_f32 %0, %8\n\t"  "v_exp_f32 %1, %9\n\t"
      "v_exp_f32 %2, %10\n\t" "v_exp_f32 %3, %11\n\t"
      "v_exp_f32 %4, %12\n\t" "v_exp_f32 %5, %13\n\t"
      "v_exp_f32 %6, %14\n\t" "v_exp_f32 %7, %15\n\t"
      "v_nop"
      : "=&v"(t0), "=&v"(t1), "=&v"(t2), "=&v"(t3),
        "=&v"(t4), "=&v"(t5), "=&v"(t6), "=&v"(t7)
      : "v"(m0), "v"(m1), "v"(m2), "v"(m3),
        "v"(m4), "v"(m5), "v"(m6), "v"(m7));
  t0 += 1.f; t1 += 1.f; t2 += 1.f; t3 += 1.f;
  t4 += 1.f; t5 += 1.f; t6 += 1.f; t7 += 1.f;
  float l0, l1, l2, l3, l4, l5, l6, l7;
  asm("v_log_f32 %0, %8\n\t"  "v_log_f32 %1, %9\n\t"
      "v_log_f32 %2, %10\n\t" "v_log_f32 %3, %11\n\t"
      "v_log_f32 %4, %12\n\t" "v_log_f32 %5, %13\n\t"
      "v_log_f32 %6, %14\n\t" "v_log_f32 %7, %15\n\t"
      "v_nop"
      : "=&v"(l0), "=&v"(l1), "=&v"(l2), "=&v"(l3),
        "=&v"(l4), "=&v"(l5), "=&v"(l6), "=&v"(l7)
      : "v"(t0), "v"(t1), "v"(t2), "v"(t3),
        "v"(t4), "v"(t5), "v"(t6), "v"(t7));
  v8f r;
  r[0] = fmaf(LN2F, l0, fmaxf(x[0], 0.f));
  r[1] = fmaf(LN2F, l1, fmaxf(x[1], 0.f));
  r[2] = fmaf(LN2F, l2, fmaxf(x[2], 0.f));
  r[3] = fmaf(LN2F, l3, fmaxf(x[3], 0.f));
  r[4] = fmaf(LN2F, l4, fmaxf(x[4], 0.f));
  r[5] = fmaf(LN2F, l5, fmaxf(x[5], 0.f));
  r[6] = fmaf(LN2F, l6, fmaxf(x[6], 0.f));
  r[7] = fmaf(LN2F, l7, fmaxf(x[7], 0.f));
  return r;
}

// ---------------- weight packing (biases pre-shifted by -ln2*colsum) --------
__device__ float pack_value(int idx,
    const float* xi, const float* Y,
    const float* wW1, const float* wb1, const float* wW2, const float* wb2,
    const float* hW, const float* hb,
    const float* gW1, const float* gb1, const float* gW2, const float* gb2,
    const float* oW, const float* ob)
{
  if (idx >= OFF_Y) {
    int r = idx - OFF_Y;
    if (r < 32)  return Y[r];
    r -= 32;
    if (r < 256) return xi[r];
    r -= 256;
    if (r < 16)  return oW[r];
    r -= 16;
    if (r == 0)  return ob[0];
    return 0.f;
  }
  int l = idx / L_STRIDE;
  int o = idx % L_STRIDE;
  if (l > 1) return 0.f;
  if (o < 64) { int k = o >> 4, n = o & 15, c = n >> 3, j = n & 7;      // BW1E
    return wW1[((l * 3 + c) * 4 + k) * 8 + j]; }
  o -= 64;
  if (o < 16) { int c = o >> 3, j = o & 7; return wb1[(l * 3 + c) * 8 + j]; } // B1E
  o -= 16;
  if (o < 256) { int k = o >> 4, n = o & 15;                            // BW2E
    if (k < 8 && n < 8)   return wW2[((l * 3 + 0) * 8 + k) * 8 + n];
    if (k >= 8 && n >= 8) return wW2[((l * 3 + 1) * 8 + (k - 8)) * 8 + (n - 8)];
    return 0.f; }
  o -= 256;
  if (o < 16) { int c = o >> 3, j = o & 7;                              // B2E
    float s = 0.f;
    for (int kk = 0; kk < 8; ++kk) s += wW2[((l * 3 + c) * 8 + kk) * 8 + j];
    return wb2[(l * 3 + c) * 8 + j] - LN2F * s; }
  o -= 16;
  if (o < 64) { int k = o >> 4, n = o & 15;                             // BW1N
    return (n < 8) ? wW1[((l * 3 + 2) * 4 + k) * 8 + n] : 0.f; }
  o -= 64;
  if (o < 16) { return (o < 8) ? wb1[(l * 3 + 2) * 8 + o] : 0.f; }      // B1N
  o -= 16;
  if (o < 256) { int k = o >> 4, n = o & 15;                            // BW2N
    return (k < 8 && n < 8) ? wW2[((l * 3 + 2) * 8 + k) * 8 + n] : 0.f; }
  o -= 256;
  if (o < 16) {                                                          // B2N
    if (o >= 8) return 0.f;
    float s = 0.f;
    for (int kk = 0; kk < 8; ++kk) s += wW2[((l * 3 + 2) * 8 + kk) * 8 + o];
    return wb2[(l * 3 + 2) * 8 + o] - LN2F * s; }
  o -= 16;
  if (o < 256) { int k = o >> 4, n = o & 15;                            // BHW (dup)
    return hW[(l * 16 + k) * 8 + (n & 7)]; }
  o -= 256;
  if (o < 16) { return hb[l * 8 + (o & 7)]; }                           // BHB (dup)
  o -= 16;
  if (o < 256) { int k = o >> 4, n = o & 15;                            // BG1
    if (k < 8 && n < 8)   return gW1[((l * 3 + 0) * 8 + k) * 8 + n];
    if (k >= 8 && n >= 8) return gW1[((l * 3 + 1) * 8 + (k - 8)) * 8 + (n - 8)];
    return 0.f; }
  o -= 256;
  if (o < 16) { int c = o >> 3, j = o & 7; return gb1[(l * 3 + c) * 8 + j]; } // BG1B
  o -= 16;
  if (o < 256) { int k = o >> 4, n = o & 15, c = k >> 3, kk = k & 7;    // BG2 vstack
    return gW2[((l * 3 + c) * 8 + kk) * 16 + n]; }
  o -= 256;
  if (o < 16) {                                                          // BG2B
    float s = 0.f;
    for (int c = 0; c < 3; ++c)
      for (int kk = 0; kk < 8; ++kk) s += gW2[((l * 3 + c) * 8 + kk) * 16 + o];
    return gb2[(l * 3 + 0) * 16 + o] + gb2[(l * 3 + 1) * 16 + o] +
           gb2[(l * 3 + 2) * 16 + o] - LN2F * s; }
  o -= 16;
  if (o < 256) { int k = o >> 4, n = o & 15;                            // BG1N
    return (k < 8 && n < 8) ? gW1[((l * 3 + 2) * 8 + k) * 8 + n] : 0.f; }
  o -= 256;
  if (o < 16) { return (o < 8) ? gb1[(l * 3 + 2) * 8 + o] : 0.f; }      // BG1NB
  o -= 16;
  if (o < 256) { int k = o >> 4, n = o & 15;                            // BG2N
    return (k < 8) ? gW2[((l * 3 + 2) * 8 + k) * 16 + n] : 0.f; }
  return 0.f;
}

extern "C" __global__ void jastrow_pack(
    const float* xi, const float* Y,
    const float* wW1, const float* wb1, const float* wW2, const float* wb2,
    const float* hW, const float* hb,
    const float* gW1, const float* gb1, const float* gW2, const float* gb2,
    const float* oW, const float* ob, float* wpack)
{
  for (int idx = blockIdx.x * blockDim.x + threadIdx.x; idx < PACK_PAD;
       idx += blockDim.x * gridDim.x)
    wpack[idx] = (idx < PACK_TOTAL)
        ? pack_value(idx, xi, Y, wW1, wb1, wW2, wb2, hW, hb, gW1, gb1, gW2, gb2, oW, ob)
        : 0.f;
}

// ---------------- main kernel: one wave32 per batch element ----------------
extern "C" __global__ __launch_bounds__(256) void jastrow_main(
    const float* __restrict__ edges_e,   // [B,16,16,4]
    const float* __restrict__ edges_n,   // [B,16,4,4]
    const float* __restrict__ wpack,     // packed weight image (L2-resident)
    float* __restrict__ out, int batch)
{
  __shared__ float sW[PACK_PAD];
  __shared__ float sScr[8][3 * SCR_REGION];   // per wave: stage | zsE | zs2

  const int tid = threadIdx.x;
  for (int idx = tid; idx < PACK_PAD; idx += 256) sW[idx] = wpack[idx];
  __syncthreads();

  const int wave = tid >> 5, lane = tid & 31;
  const int half = lane >> 4, n = lane & 15, m = n;
  float* S   = &sScr[wave][0];
  float* Szs = &sScr[wave][SCR_REGION];
  float* Sz2 = &sScr[wave][2 * SCR_REGION];

  float yreg[4];
#pragma unroll
  for (int a = 0; a < 4; ++a) yreg[a] = sW[OFF_Y + a * 8 + (n & 7)];
  const float own = sW[OFF_OW + n];
  const float obias = sW[OFF_OB];

  // per-lane einsum constants: a lane's 8 rows j all have spin sj == half.
  // gate(i) = (si == half) XOR ch  ->  (si == (half^ch)); diagonal j==i only
  // matters on same-spin lanes of the matching half.
  const int ch = n >> 3;
  const int hc = half ^ ch;
  const float gA = (hc == 0) ? 1.f : 0.f;            // gate for i in [0,8)
  const float gB = 1.f - gA;                         // gate for i in [8,16)
  const float dA = (ch == 0 && half == 0) ? 1.f : 0.f;
  const float dB = (ch == 0 && half == 1) ? 1.f : 0.f;

  for (int b = blockIdx.x * 8 + wave; b < batch; b += gridDim.x * 8) {
    const float* eE = edges_e + (size_t)b * 1024;
    const float* eN = edges_n + (size_t)b * 256;

    // early prefetch of this element's edges (global_prefetch_b8)
    __builtin_prefetch(eE + lane * 32, 0, 3);
    if (lane < 8) __builtin_prefetch(eN + lane * 32, 0, 3);

    // x in D-layout: lane(half,n) holds rows v+8*half, col n
    v8f x;
#pragma unroll
    for (int v = 0; v < 8; ++v) x[v] = sW[OFF_XINIT + (v + 8 * half) * 16 + n];

#pragma unroll 1
    for (int l = 0; l < 2; ++l) {
      const float* W = sW + l * L_STRIDE;
      const v2f bw1e = load_b4(W + OFF_BW1E, 0, half, n);
      const v2f bw1n = load_b4(W + OFF_BW1N, 0, half, n);
      // hoist loop-invariant stage-2 B fragments into registers
      const v2f bw2e0 = load_b4(W + OFF_BW2E, 0,  half, n);
      const v2f bw2e1 = load_b4(W + OFF_BW2E, 4,  half, n);
      const v2f bw2e2 = load_b4(W + OFF_BW2E, 8,  half, n);
      const v2f bw2e3 = load_b4(W + OFF_BW2E, 12, half, n);
      const v2f bw2n0 = load_b4(W + OFF_BW2N, 0,  half, n);
      const v2f bw2n1 = load_b4(W + OFF_BW2N, 4,  half, n);
      const float b1e = W[OFF_B1E + n], b2e = W[OFF_B2E + n];
      const float b1n = W[OFF_B1N + n], b2n = W[OFF_B2N + n];

      // ---- hx = x @ hW + hb  (hW cols duplicated -> D col n == hx[j, n&7]) ----
      store_d(S, x, half, n);
      v8f hx = bias8(W[OFF_BHB + n]);
#pragma unroll
      for (int k = 0; k < 16; k += 4)
        hx = wmma4(load_a4(S, k, half, m), load_b4(W + OFF_BHW, k, half, n), hx);

      // ---- electron filter channels + masked einsum over j ----
#pragma unroll 1
      for (int ih = 0; ih < 2; ++ih) {               // i = ih*8 + u
        const float g  = ih ? gB : gA;
        const float dX = ih ? dB : dA;
        const float* eEh = eE + ih * 512;
        float* zrow = Szs + ih * 8 * SCRW;
#pragma unroll
        for (int u = 0; u < 8; ++u) {
          // A = edges_elec[b,i,:,:] (16 rows j x 4 basis), straight from global
          v2f a = *reinterpret_cast<const v2f*>(eEh + (u * 16 + m) * 4 + 2 * half);
          v8f c1 = wmma4(a, bw1e, bias8(b1e));       // both spin channels at once
          c1 = sp8(c1);
          store_d(S, c1, half, n);                   // D-layout -> A-layout via LDS
          v8f c2 = bias8(b2e);
          c2 = wmma4(load_a4(S, 0,  half, m), bw2e0, c2);
          c2 = wmma4(load_a4(S, 4,  half, m), bw2e1, c2);
          c2 = wmma4(load_a4(S, 8,  half, m), bw2e2, c2);
          c2 = wmma4(load_a4(S, 12, half, m), bw2e3, c2);
          // zs[ch][i, n&7] = gate*sum_j w*hx  - (diagonal j==i term)
          float dot = 0.f;
#pragma unroll
          for (int v = 0; v < 8; ++v) dot = fmaf(c2[v], hx[v], dot);
          float pe = c2[u] * hx[u];                  // static index: j == i row
          float part = fmaf(g, dot, -(dX * pe));
          part += __shfl_xor(part, 16, 32);          // add partner half's rows
          zrow[u * SCRW + n] = part;                 // both halves: same value
        }
      }

      // ---- nuclear channel: rows are (i,a) pairs, 4 tiles of 16 ----
#pragma unroll 1
      for (int t = 0; t < 4; ++t) {
        v2f a = *reinterpret_cast<const v2f*>(eN + (t * 16 + m) * 4 + 2 * half);
        v8f c1 = wmma4(a, bw1n, bias8(b1n));
        c1 = sp8(c1);
        store_d(S, c1, half, n);
        v8f c2 = bias8(b2n);
        c2 = wmma4(load_a4(S, 0, half, m), bw2n0, c2);
        c2 = wmma4(load_a4(S, 4, half, m), bw2n1, c2);
        // contract over atoms a (groups of 4 rows) against Y[a, n&7]
        float s0 = fmaf(c2[0], yreg[0], fmaf(c2[1], yreg[1],
                   fmaf(c2[2], yreg[2], c2[3] * yreg[3])));
        float s1 = fmaf(c2[4], yreg[0], fmaf(c2[5], yreg[1],
                   fmaf(c2[6], yreg[2], c2[7] * yreg[3])));
        int i0 = t * 4 + 2 * half;
        Sz2[i0 * SCRW + n]       = s0;
        Sz2[(i0 + 1) * SCRW + n] = s1;
      }

      // ---- g-subnet, elec channels combined (blockdiag gW1, vstack gW2) ----
      v8f g1 = bias8(W[OFF_BG1B + n]);
#pragma unroll
      for (int k = 0; k < 16; k += 4)
        g1 = wmma4(load_a4(Szs, k, half, m), load_b4(W + OFF_BG1, k, half, n), g1);
      g1 = sp8(g1);
      store_d(S, g1, half, n);
      {
        float gb = W[OFF_BG2B + n];                  // summed+shifted gb2
#pragma unroll
        for (int v = 0; v < 8; ++v) x[v] += gb;
      }
#pragma unroll
      for (int k = 0; k < 16; k += 4)                // residual accumulated into x
        x = wmma4(load_a4(S, k, half, m), load_b4(W + OFF_BG2, k, half, n), x);

      // ---- g-subnet, nuclear channel (K=8, zero-padded, accumulates into x) ----
      v8f g1n = bias8(W[OFF_BG1NB + n]);
#pragma unroll
      for (int k = 0; k < 8; k += 4)
        g1n = wmma4(load_a4(Sz2, k, half, m), load_b4(W + OFF_BG1N, k, half, n), g1n);
      g1n = sp8(g1n);
      store_d(S, g1n, half, n);
#pragma unroll
      for (int k = 0; k < 8; k += 4)
        x = wmma4(load_a4(S, k, half, m), load_b4(W + OFF_BG2N, k, half, n), x);
    }

    // ---- readout: sum_e sum_elec x[elec,e]*oW[e] + 16*ob ----
    float acc = 0.f;
#pragma unroll
    for (int v = 0; v < 8; ++v) acc = fmaf(x[v], own, acc);
#pragma unroll
    for (int off = 16; off >= 1; off >>= 1) acc += __shfl_xor(acc, off, 32);
    if (lane == 0) out[b] = acc + 16.f * obias;
  }
}

// ---------------- host entry ----------------
extern "C" void kernel_launch(void* const* d_in, const int* in_sizes, int n_in,
                              void* d_out, int out_size, void* d_ws, size_t ws_size,
                              hipStream_t stream) {
  const float* edges_e = (const float*)d_in[0];
  const float* edges_n = (const float*)d_in[1];
  const float* x_init  = (const float*)d_in[2];
  const float* Y       = (const float*)d_in[3];
  const float* wW1     = (const float*)d_in[4];
  const float* wb1     = (const float*)d_in[5];
  const float* wW2     = (const float*)d_in[6];
  const float* wb2     = (const float*)d_in[7];
  const float* hW      = (const float*)d_in[8];
  const float* hb      = (const float*)d_in[9];
  const float* gW1     = (const float*)d_in[10];
  const float* gb1     = (const float*)d_in[11];
  const float* gW2     = (const float*)d_in[12];
  const float* gb2     = (const float*)d_in[13];
  const float* oW      = (const float*)d_in[14];
  const float* ob      = (const float*)d_in[15];
  float* wpack = (float*)d_ws;
  float* outp  = (float*)d_out;

  int batch = in_sizes[0] / (16 * 16 * 4);

  jastrow_pack<<<dim3(4), dim3(256), 0, stream>>>(
      x_init, Y, wW1, wb1, wW2, wb2, hW, hb, gW1, gb1, gW2, gb2, oW, ob, wpack);

  int blocks = (batch + 7) / 8;     // one batch element per wave32
  if (blocks < 1) blocks = 1;
  jastrow_main<<<dim3(blocks), dim3(256), 0, stream>>>(
      edges_e, edges_n, wpack, outp, batch);

  (void)n_in; (void)out_size; (void)ws_size;
}